// ScaledDotProductAttention_23545010717013
// MI455X (gfx1250) — compile-verified
//
#include <hip/hip_runtime.h>

// Scaled dot-product attention with full weights output, fp32, for MI455X.
// Uses V_WMMA_F32_16X16X4_F32 (CDNA5 wave32 matrix op) for both QK^T and PV.
//
// Roofline: must write 256MB weights + read 256MB mask + ~50MB QKV ≈ 25us at
// 23.3 TB/s; fp32 compute ~34 GFLOP on the f32 WMMA pipe keeps reference
// precision. Scores staged in LDS (128KB of the 320KB/WGP) so the weights
// matrix touches HBM exactly once.
//
// v2: QK^T uses 4 independent WMMA accumulator chains (dep distance 4) so the
// matrix pipe issues back-to-back instead of stalling on the accumulator RAW
// hazard (v1 asm showed s_delay_alu VALU_DEP_1 between every wmma pair).

typedef __attribute__((ext_vector_type(2))) float v2f;
typedef __attribute__((ext_vector_type(8))) float v8f;

#define SLEN 2048
#define DK   128
#define SROW 2052          // padded LDS row stride (floats): kills bank conflicts
#define MASK_VALUE -10000000.0f
#define RSQRT_DK 0.08838834764831845f   // 1/sqrt(128)

__global__ __launch_bounds__(256)
void sdpa_f32_wmma_kernel(const float* __restrict__ Q,
                          const float* __restrict__ Kmat,
                          const float* __restrict__ V,
                          const int*   __restrict__ Mask,
                          float* __restrict__ attn,    // [16,2048,128]
                          float* __restrict__ wout)    // [16,2048,2048]
{
    extern __shared__ float smem[];
    float* sc     = smem;                  // [16][SROW] scores -> exp values
    float* oacc   = sc + 16 * SROW;        // [16][128] O accumulator
    float* pred   = oacc + 2048;           // [8][16] per-wave partials
    float* rowmax = pred + 128;            // [16]
    float* rinv   = rowmax + 16;           // [16]

    const int tid  = threadIdx.x;
    const int wave = tid >> 5;
    const int lane = tid & 31;
    const int half = lane >> 4;            // 0: lanes 0-15, 1: lanes 16-31
    const int m    = lane & 15;

    const int batch = blockIdx.x >> 7;     // 16 batches
    const int q0    = (blockIdx.x & 127) << 4;  // 128 query tiles of 16

    // ---- init LDS ----
    for (int i = tid; i < 2048; i += 256) oacc[i] = 0.0f;
    if (tid < 16) rowmax[tid] = -3.0e38f;
    __syncthreads();

    // ---- preload Q tile: lane holds Q[q0+m][4*i + 2*half .. +1] ----
    const float* qrow = Q + (size_t)(batch * SLEN + q0 + m) * DK;
    float qa[64];
#pragma unroll
    for (int i = 0; i < 32; ++i) {
        float2 t = *(const float2*)(qrow + 4 * i + 2 * half);
        qa[2 * i] = t.x; qa[2 * i + 1] = t.y;
    }

    // ---- Phase A: masked scores for this wave's 16 key tiles ----
    const float* kbase = Kmat + (size_t)batch * SLEN * DK;
    float lmax[8];
#pragma unroll
    for (int j = 0; j < 8; ++j) lmax[j] = -3.0e38f;

    for (int t = 0; t < 16; ++t) {
        const int kt = wave * 16 + t;          // key tile index
        const int n  = kt * 16 + m;            // this lane's key row for B
        const float* kr = kbase + (size_t)n * DK;

        // 4 independent accumulator chains -> no WMMA->WMMA RAW stall
        v8f c0 = {}, c1 = {}, c2 = {}, c3 = {};
#pragma unroll
        for (int i = 0; i < 32; i += 4) {      // K = 128 in steps of 4x4
            v2f a0; a0.x = qa[2*i+0]; a0.y = qa[2*i+1];
            v2f a1; a1.x = qa[2*i+2]; a1.y = qa[2*i+3];
            v2f a2; a2.x = qa[2*i+4]; a2.y = qa[2*i+5];
            v2f a3; a3.x = qa[2*i+6]; a3.y = qa[2*i+7];
            float2 k0 = *(const float2*)(kr + 4*(i+0) + 2*half);
            float2 k1 = *(const float2*)(kr + 4*(i+1) + 2*half);
            float2 k2 = *(const float2*)(kr + 4*(i+2) + 2*half);
            float2 k3 = *(const float2*)(kr + 4*(i+3) + 2*half);
            v2f b0; b0.x = k0.x; b0.y = k0.y;
            v2f b1; b1.x = k1.x; b1.y = k1.y;
            v2f b2; b2.x = k2.x; b2.y = k2.y;
            v2f b3; b3.x = k3.x; b3.y = k3.y;
            c0 = __builtin_amdgcn_wmma_f32_16x16x4_f32(false, a0, false, b0, (short)0, c0, false, false);
            c1 = __builtin_amdgcn_wmma_f32_16x16x4_f32(false, a1, false, b1, (short)0, c1, false, false);
            c2 = __builtin_amdgcn_wmma_f32_16x16x4_f32(false, a2, false, b2, (short)0, c2, false, false);
            c3 = __builtin_amdgcn_wmma_f32_16x16x4_f32(false, a3, false, b3, (short)0, c3, false, false);
        }

        // c[j] = score[row j+8*half][col kt*16+m]
        const int* mrow = Mask + (size_t)(batch * SLEN + q0) * SLEN + kt * 16 + m;
#pragma unroll
        for (int j = 0; j < 8; ++j) {
            const int r = j + 8 * half;
            float s = ((c0[j] + c1[j]) + (c2[j] + c3[j])) * RSQRT_DK;
            const int mk = mrow[(size_t)r * SLEN];
            s = mk ? MASK_VALUE : s;           // score*(1-m) + m*MASK_VALUE
            sc[r * SROW + kt * 16 + m] = s;
            lmax[j] = fmaxf(lmax[j], s);
        }
    }

    // row-max: reduce over the 16 lanes of each half, then across waves
#pragma unroll
    for (int j = 0; j < 8; ++j) {
        float v = lmax[j];
        v = fmaxf(v, __shfl_xor(v, 1, 32));
        v = fmaxf(v, __shfl_xor(v, 2, 32));
        v = fmaxf(v, __shfl_xor(v, 4, 32));
        v = fmaxf(v, __shfl_xor(v, 8, 32));
        if (m == 0) pred[wave * 16 + j + 8 * half] = v;
    }
    __syncthreads();
    if (tid < 16) {
        float v = pred[tid];
#pragma unroll
        for (int w = 1; w < 8; ++w) v = fmaxf(v, pred[w * 16 + tid]);
        rowmax[tid] = v;
    }
    __syncthreads();

    // ---- Phase B: e = exp(s - rowmax), row sums (wave owns cols [wave*256,+256)) ----
    for (int r = 0; r < 16; ++r) {
        const float rm = rowmax[r];
        float* srow = sc + r * SROW + wave * 256;
        float ps = 0.0f;
#pragma unroll
        for (int c8 = 0; c8 < 8; ++c8) {
            const int col = c8 * 32 + lane;
            float s = srow[col];
            float e = (s < -1.0e6f) ? 0.0f : __expf(s - rm);  // exact 0 when masked
            srow[col] = e;
            ps += e;
        }
        ps += __shfl_xor(ps, 1, 32);
        ps += __shfl_xor(ps, 2, 32);
        ps += __shfl_xor(ps, 4, 32);
        ps += __shfl_xor(ps, 8, 32);
        ps += __shfl_xor(ps, 16, 32);
        if (lane == 0) pred[wave * 16 + r] = ps;
    }
    __syncthreads();
    if (tid < 16) {
        float v = 0.0f;
#pragma unroll
        for (int w = 0; w < 8; ++w) v += pred[w * 16 + tid];
        rinv[tid] = 1.0f / v;
    }
    __syncthreads();

    // ---- Phase C1: stream normalized weights to global (float4, coalesced) ----
    float* wbase = wout + (size_t)(batch * SLEN + q0) * SLEN;
    for (int r = 0; r < 16; ++r) {
        const float ri = rinv[r];
#pragma unroll
        for (int cb = 0; cb < 2; ++cb) {
            const int i4 = cb * 256 + tid;     // 512 float4 per row
            float4 e4 = *(const float4*)(sc + r * SROW + i4 * 4);
            float4 w4 = make_float4(e4.x * ri, e4.y * ri, e4.z * ri, e4.w * ri);
            *(float4*)(wbase + (size_t)r * SLEN + i4 * 4) = w4;
        }
    }

    // ---- Phase C2: O_partial = E_chunk * V_chunk via f32 WMMA ----
    // 8 independent accumulators (dep distance 8) -> no RAW stalls here.
    const float* vbase = V + (size_t)batch * SLEN * DK;
    v8f o[8];
#pragma unroll
    for (int nt = 0; nt < 8; ++nt) o[nt] = (v8f){};

    for (int ks = 0; ks < 64; ++ks) {          // 256 keys of this wave, K-step 4
        const int k0 = wave * 256 + ks * 4;
        float2 av = *(const float2*)(sc + m * SROW + k0 + 2 * half); // ds_load_b64
        v2f a; a.x = av.x; a.y = av.y;
        const float* v0 = vbase + (size_t)(k0 + 2 * half) * DK + m;
#pragma unroll
        for (int nt = 0; nt < 8; ++nt) {       // 8 output-dim tiles of 16
            v2f b; b.x = v0[nt * 16]; b.y = v0[nt * 16 + DK];
            o[nt] = __builtin_amdgcn_wmma_f32_16x16x4_f32(
                        false, a, false, b, (short)0, o[nt], false, false);
        }
    }

    // reduce partial O across waves in LDS (ds_add_f32)
#pragma unroll
    for (int nt = 0; nt < 8; ++nt)
#pragma unroll
        for (int j = 0; j < 8; ++j)
            atomicAdd(&oacc[(j + 8 * half) * DK + nt * 16 + m], o[nt][j]);
    __syncthreads();

    // ---- write attention = O * (1/rowsum), float4 coalesced ----
    float* abase = attn + (size_t)(batch * SLEN + q0) * DK;
#pragma unroll
    for (int i = 0; i < 2; ++i) {
        const int i4  = i * 256 + tid;         // 512 float4 total
        const int r   = (i4 * 4) >> 7;
        const int d   = (i4 * 4) & 127;
        const float ri = rinv[r];
        float4 t = *(const float4*)(oacc + i4 * 4);
        float4 ov = make_float4(t.x * ri, t.y * ri, t.z * ri, t.w * ri);
        *(float4*)(abase + (size_t)r * DK + d) = ov;
    }
}

extern "C" void kernel_launch(void* const* d_in, const int* in_sizes, int n_in,
                              void* d_out, int out_size, void* d_ws, size_t ws_size,
                              hipStream_t stream) {
    const float* Q    = (const float*)d_in[0];
    const float* Kmat = (const float*)d_in[1];
    const float* V    = (const float*)d_in[2];
    const int*   Mask = (const int*)d_in[3];

    float* attn = (float*)d_out;                       // [16,2048,128] first
    float* wout = attn + (size_t)16 * SLEN * DK;       // then [16,2048,2048]

    const int    blocks = 16 * (SLEN / 16);            // 2048: (batch, q-tile)
    const size_t smem   = (size_t)(16 * SROW + 2048 + 128 + 16 + 16) * sizeof(float);

    sdpa_f32_wmma_kernel<<<blocks, 256, smem, stream>>>(Q, Kmat, V, Mask, attn, wout);
}